// VariationalSDE_5669356834541
// MI455X (gfx1250) — compile-verified
//
#include <hip/hip_runtime.h>
#include <hip/hip_bf16.h>
#include <math.h>

// ---- problem constants (from reference) ----
#define N_T      100
#define NSTEPS   (N_T - 1)                 // 99 EM steps
#define DT       (3.0f / 99.0f)            // linspace(0,3,100) -> uniform dt
#define SQRT_DT  0.17407765595569785f      // sqrt(3/99)
#define DIFF     1.4142135623730951f       // sqrt(2)
#define GDT      (DIFF * SQRT_DT)          // diff_coef * sqrt(dt)
#define KC       (DT * 0.25f)              // dt * 0.5 * (1/g^2),  g^2 = 2

#if __has_builtin(__builtin_amdgcn_tanhf)
#define FAST_TANH(x) __builtin_amdgcn_tanhf(x)   // v_tanh_f32 (1 trans op)
#else
#define FAST_TANH(x) tanhf(x)
#endif

#define BLOCK 256

// async global -> LDS staging of one dW element per lane
__device__ __forceinline__ void async_dw(unsigned lds_byte_off, const float* gaddr) {
    unsigned long long ga = (unsigned long long)(const void*)gaddr;
    asm volatile("global_load_async_to_lds_b32 %0, %1, off"
                 :: "v"(lds_byte_off), "v"(ga)
                 : "memory");
}

__launch_bounds__(BLOCK, 4)
__global__ void sde_em_kernel(const float* __restrict__ z0_raw,
                              const float* __restrict__ dW,    // [99][n]
                              const float* __restrict__ W1,    // [8]
                              const float* __restrict__ b1,    // [8]
                              const float* __restrict__ wt,    // [8]
                              const float* __restrict__ W2,    // [8]
                              const float* __restrict__ b2,    // [1]
                              float* __restrict__ out,         // z_t [100][n] ++ kld [n]
                              int n) {
    __shared__ float s_dw[2][BLOCK];
    const int tid = threadIdx.x;
    const int j   = blockIdx.x * BLOCK + tid;
    if (j >= n) return;

    // hoist the tiny MLP weights into registers (uniform across all lanes)
    float w1r[8], b1r[8], wtr[8], w2r[8];
#pragma unroll
    for (int k = 0; k < 8; ++k) {
        w1r[k] = W1[k]; b1r[k] = b1[k]; wtr[k] = wt[k]; w2r[k] = W2[k];
    }
    const float b2r = b2[0];

    // LDS byte offsets for this lane's two staging slots
    // (low 32 bits of a flat shared pointer == LDS byte offset)
    const unsigned lds0 = (unsigned)(size_t)&s_dw[0][tid];
    const unsigned lds1 = (unsigned)(size_t)&s_dw[1][tid];

    float z   = z0_raw[j] * 3.0f;
    float kld = 0.0f;
    __builtin_nontemporal_store(z, out + j);          // z_t[0]

    // prime the pipeline: stage dW row 0 into buffer 0
    async_dw(lds0, dW + j);

#pragma unroll 1
    for (int i = 0; i < NSTEPS; ++i) {
        if (i + 1 < NSTEPS) {
            // stage next step's dW while we compute this step
            async_dw(((i + 1) & 1) ? lds1 : lds0,
                     dW + (size_t)(i + 1) * n + j);
            // async loads complete in order: <=1 outstanding => row i is resident
            asm volatile("s_wait_asynccnt 1" ::: "memory");
        } else {
            asm volatile("s_wait_asynccnt 0" ::: "memory");
        }
        const float dw = s_dw[i & 1][tid];

        // time-conditioned drift MLP: f = W2 . tanh(z*W1 + t0*wt + b1) + b2
        const float t0 = (float)i * DT;
        float f = b2r;
#pragma unroll
        for (int k = 0; k < 8; ++k) {
            float h = FAST_TANH(fmaf(z, w1r[k], fmaf(t0, wtr[k], b1r[k])));
            f = fmaf(h, w2r[k], f);
        }

        // Euler-Maruyama update: z += f*dt + g*sqrt(dt)*dw
        z = fmaf(f, DT, fmaf(GDT, dw, z));

        // KL term: u = (f + 0.1*z_new)/g ; kld += dt*0.5*u^2  ==  (f+0.1z)^2 * dt/4
        const float s = fmaf(0.1f, z, f);
        kld = fmaf(s * s, KC, kld);

        __builtin_nontemporal_store(z, out + (size_t)(i + 1) * n + j);
    }

    __builtin_nontemporal_store(kld, out + (size_t)N_T * n + j);
}

extern "C" void kernel_launch(void* const* d_in, const int* in_sizes, int n_in,
                              void* d_out, int out_size, void* d_ws, size_t ws_size,
                              hipStream_t stream) {
    const float* z0 = (const float*)d_in[0];
    const float* dW = (const float*)d_in[1];
    const float* W1 = (const float*)d_in[2];
    const float* b1 = (const float*)d_in[3];
    const float* wt = (const float*)d_in[4];
    const float* W2 = (const float*)d_in[5];
    const float* b2 = (const float*)d_in[6];
    const int n = in_sizes[0];                 // 262144 samples (z0_raw flat count)

    dim3 block(BLOCK);
    dim3 grid((n + BLOCK - 1) / BLOCK);
    hipLaunchKernelGGL(sde_em_kernel, grid, block, 0, stream,
                       z0, dW, W1, b1, wt, W2, b2, (float*)d_out, n);
}